// PlainVoxels_71210557768318
// MI455X (gfx1250) — compile-verified
//
#include <hip/hip_runtime.h>

typedef __attribute__((ext_vector_type(2))) float v2f;
typedef __attribute__((ext_vector_type(8))) float v8f;

#define NRAYS 4096
#define NSAMP 128
#define GDIM  160
#define RAYS_PER_BLOCK 8

__global__ __launch_bounds__(256) void plainvoxels_kernel(
    const float* __restrict__ rays_o,
    const float* __restrict__ rays_d,
    const float* __restrict__ rays_d_norm,
    const float* __restrict__ t_nears,
    const float* __restrict__ t_fars,
    const float* __restrict__ grid,
    const float* __restrict__ beta,
    float* __restrict__ out)
{
    __shared__ float red[RAYS_PER_BLOCK * 32 * 8];

    const int lane = threadIdx.x & 31;
    const int wave = threadIdx.x >> 5;
    const int ray  = blockIdx.x * RAYS_PER_BLOCK + wave;

    float* out_rgb    = out;                 // [4096,3]
    float* out_depth  = out + NRAYS * 3;     // [4096,1]
    float* out_normal = out + NRAYS * 4;     // [4096,3]
    float* out_acc    = out + NRAYS * 7;     // [4096,1]
    float* out_grads  = out + NRAYS * 8;     // [524288,3]

    const float ox = rays_o[ray * 3 + 0];
    const float oy = rays_o[ray * 3 + 1];
    const float oz = rays_o[ray * 3 + 2];
    const float dx = rays_d[ray * 3 + 0];
    const float dy = rays_d[ray * 3 + 1];
    const float dz = rays_d[ray * 3 + 2];

    const float beta_c = fmaxf(beta[0], 1e-4f);
    const float alpha  = 1.0f / beta_c;
    const float UB     = (float)(159.0 - 0.0001);   // G - 1 - 1e-4

    float taus[4], tmids[4];
    float rgbv[4][3], nrmv[4][3];

    // ---- Phase 1: per-sample trilerp, sdf grad, density ----
    #pragma unroll
    for (int s = 0; s < 4; ++s) {
        const int j = lane * 4 + s;          // sample within ray (contiguous chunk)
        const int m = ray * NSAMP + j;       // global sample index

        const float tn   = t_nears[m];
        const float tf   = t_fars[m];
        const float tmid = 0.5f * (tn + tf);
        const float dt   = tf - tn;

        const float px = ox + tmid * dx;
        const float py = oy + tmid * dy;
        const float pz = oz + tmid * dz;

        const float uxr = px / 0.01f;
        const float uyr = py / 0.01f;
        const float uzr = pz / 0.01f;
        const float ax = (uxr > 0.0f && uxr < UB) ? 1.0f : 0.0f;  // clip gradient
        const float ay = (uyr > 0.0f && uyr < UB) ? 1.0f : 0.0f;
        const float az = (uzr > 0.0f && uzr < UB) ? 1.0f : 0.0f;

        const float ux = fminf(fmaxf(uxr, 0.0f), UB);
        const float uy = fminf(fmaxf(uyr, 0.0f), UB);
        const float uz = fminf(fmaxf(uzr, 0.0f), UB);

        const int ix0 = (int)ux;   // u >= 0 -> trunc == floor
        const int iy0 = (int)uy;
        const int iz0 = (int)uz;
        const float fx = ux - (float)ix0;
        const float fy = uy - (float)iy0;
        const float fz = uz - (float)iz0;
        int ix1 = ix0 + 1; if (ix1 > GDIM - 1) ix1 = GDIM - 1;
        int iy1 = iy0 + 1; if (iy1 > GDIM - 1) iy1 = GDIM - 1;
        int iz1 = iz0 + 1; if (iz1 > GDIM - 1) iz1 = GDIM - 1;

        const float gx = 1.0f - fx, gy = 1.0f - fy, gz = 1.0f - fz;

        const float* c000 = grid + (size_t)((ix0 * GDIM + iy0) * GDIM + iz0) * 5;
        const float* c001 = grid + (size_t)((ix0 * GDIM + iy0) * GDIM + iz1) * 5;
        const float* c010 = grid + (size_t)((ix0 * GDIM + iy1) * GDIM + iz0) * 5;
        const float* c011 = grid + (size_t)((ix0 * GDIM + iy1) * GDIM + iz1) * 5;
        const float* c100 = grid + (size_t)((ix1 * GDIM + iy0) * GDIM + iz0) * 5;
        const float* c101 = grid + (size_t)((ix1 * GDIM + iy0) * GDIM + iz1) * 5;
        const float* c110 = grid + (size_t)((ix1 * GDIM + iy1) * GDIM + iz0) * 5;
        const float* c111 = grid + (size_t)((ix1 * GDIM + iy1) * GDIM + iz1) * 5;

        const float w000 = gx * gy * gz, w001 = gx * gy * fz;
        const float w010 = gx * fy * gz, w011 = gx * fy * fz;
        const float w100 = fx * gy * gz, w101 = fx * gy * fz;
        const float w110 = fx * fy * gz, w111 = fx * fy * fz;

        float emb[5];
        #pragma unroll
        for (int c = 0; c < 5; ++c) {
            emb[c] = c000[c] * w000 + c001[c] * w001 + c010[c] * w010 + c011[c] * w011
                   + c100[c] * w100 + c101[c] * w101 + c110[c] * w110 + c111[c] * w111;
        }

        // analytic gradient of sdf (channel 0) wrt world x
        const float s000 = c000[0], s001 = c001[0], s010 = c010[0], s011 = c011[0];
        const float s100 = c100[0], s101 = c101[0], s110 = c110[0], s111 = c111[0];
        const float dux = (s100 - s000) * gy * gz + (s101 - s001) * gy * fz
                        + (s110 - s010) * fy * gz + (s111 - s011) * fy * fz;
        const float duy = (s010 - s000) * gx * gz + (s011 - s001) * gx * fz
                        + (s110 - s100) * fx * gz + (s111 - s101) * fx * fz;
        const float duz = (s001 - s000) * gx * gy + (s011 - s010) * gx * fy
                        + (s101 - s100) * fx * gy + (s111 - s110) * fx * fy;
        const float gxv = (dux / 0.01f) * ax;
        const float gyv = (duy / 0.01f) * ay;
        const float gzv = (duz / 0.01f) * az;

        const float sdf = emb[0];
        const float vw  = emb[4];
        const bool mask = (vw >= 1.0f);

        const float nl  = sqrtf(gxv * gxv + gyv * gyv + gzv * gzv);
        const float inv = 1.0f / fmaxf(nl, 1e-12f);
        nrmv[s][0] = gxv * inv; nrmv[s][1] = gyv * inv; nrmv[s][2] = gzv * inv;
        rgbv[s][0] = emb[1];    rgbv[s][1] = emb[2];    rgbv[s][2] = emb[3];
        tmids[s]   = tmid;

        const float sgn   = (sdf > 0.0f) ? 1.0f : ((sdf < 0.0f) ? -1.0f : 0.0f);
        const float sigma = 0.5f * alpha * (1.0f + sgn * expm1f(-fabsf(sdf) / beta_c));
        taus[s] = mask ? (sigma * dt) : 0.0f;

        out_grads[(size_t)m * 3 + 0] = mask ? gxv : 0.0f;
        out_grads[(size_t)m * 3 + 1] = mask ? gyv : 0.0f;
        out_grads[(size_t)m * 3 + 2] = mask ? gzv : 0.0f;
    }

    // ---- Phase 2: wave-parallel exclusive scan of tau, weights, partial sums ----
    const float p0 = taus[0];
    const float p1 = p0 + taus[1];
    const float p2 = p1 + taus[2];
    const float p3 = p2 + taus[3];

    float incl = p3;
    #pragma unroll
    for (int off = 1; off < 32; off <<= 1) {
        float nb = __shfl_up(incl, off, 32);
        if (lane >= off) incl += nb;
    }
    const float excl = incl - p3;   // exclusive cum before this lane's chunk

    float acc[8] = {0.0f, 0.0f, 0.0f, 0.0f, 0.0f, 0.0f, 0.0f, 0.0f};
    const float pref[4] = {0.0f, p0, p1, p2};
    #pragma unroll
    for (int s = 0; s < 4; ++s) {
        const float e = excl + pref[s];                 // exclusive cum before sample
        const float w = expf(-e) * (-expm1f(-taus[s])); // trans * (1 - exp(-tau))
        acc[0] += w * rgbv[s][0];
        acc[1] += w * rgbv[s][1];
        acc[2] += w * rgbv[s][2];
        acc[3] += w * tmids[s];
        acc[4] += w * nrmv[s][0];
        acc[5] += w * nrmv[s][1];
        acc[6] += w * nrmv[s][2];
        acc[7] += w;
    }

    // ---- Phase 3: cross-lane reduction via V_WMMA_F32_16X16X4_F32 ----
    // Stage per-lane partials in LDS, then D = ones(16x4) x B(4x16) + C
    // gives column sums of B; 4 chained WMMAs cover all 32 lanes.
    float* sl = red + wave * 256;
    #pragma unroll
    for (int f = 0; f < 8; ++f) sl[lane * 8 + f] = acc[f];
    __syncthreads();

    const int h    = lane >> 4;      // half-wave
    const int n    = lane & 15;      // column index
    const int gsel = (n >> 3) & 1;   // lane-group-of-4 select
    const int fcol = n & 7;          // feature of this column

    v2f a; a[0] = 1.0f; a[1] = 1.0f;            // A = all ones (16x4)
    v8f c = {0.0f, 0.0f, 0.0f, 0.0f, 0.0f, 0.0f, 0.0f, 0.0f};
    #pragma unroll
    for (int t = 0; t < 4; ++t) {
        const int l0 = t * 8 + gsel * 4 + h * 2;            // source lanes l0, l0+1
        v2f b;
        b[0] = sl[l0 * 8 + fcol];
        b[1] = sl[(l0 + 1) * 8 + fcol];
        // D[m][n] = sum_k B[k][n] + C[m][n]  (A == ones)
        c = __builtin_amdgcn_wmma_f32_16x16x4_f32(
            false, a, false, b, (short)0, c, false, false);
    }
    // c[0] at lane L = column-sum of column (L & 15); pair halves to total 32 lanes
    const float col = c[0];
    const float tot = col + __shfl_xor(col, 8, 32);

    if (lane < 8) {
        if (lane < 3)       out_rgb[ray * 3 + lane]        = tot;
        else if (lane == 3) out_depth[ray]                 = tot / rays_d_norm[ray];
        else if (lane < 7)  out_normal[ray * 3 + (lane-4)] = tot;
        else                out_acc[ray]                   = tot;
    }
}

extern "C" void kernel_launch(void* const* d_in, const int* in_sizes, int n_in,
                              void* d_out, int out_size, void* d_ws, size_t ws_size,
                              hipStream_t stream) {
    const float* rays_o      = (const float*)d_in[0];
    const float* rays_d      = (const float*)d_in[1];
    const float* rays_d_norm = (const float*)d_in[2];
    // d_in[3] = ray_indices (int32): implied by sample index, unused
    const float* t_nears     = (const float*)d_in[4];
    const float* t_fars      = (const float*)d_in[5];
    const float* grid        = (const float*)d_in[6];
    const float* beta        = (const float*)d_in[7];

    plainvoxels_kernel<<<NRAYS / RAYS_PER_BLOCK, 256, 0, stream>>>(
        rays_o, rays_d, rays_d_norm, t_nears, t_fars, grid, beta, (float*)d_out);
}